// HistogramLayer_81939386073088
// MI455X (gfx1250) — compile-verified
//
#include <hip/hip_runtime.h>
#include <cstdint>

#define NN 131072
#define DD 512
#define NB 64

// ---- monotone uint key for order-independent float max via atomicMax(uint) ----
__device__ __forceinline__ uint32_t fkey(float x) {
    uint32_t u = __float_as_uint(x);
    return (u & 0x80000000u) ? ~u : (u | 0x80000000u);
}
__device__ __forceinline__ float funkey(uint32_t k) {
    return (k & 0x80000000u) ? __uint_as_float(k & 0x7fffffffu) : __uint_as_float(~k);
}

// ---- K0: log_probs[b][d] = log(freq[b][d]) - log(sum_b freq[b][d]); init scalars ----
__global__ void k_logp(const float* __restrict__ freq, float* __restrict__ logp,
                       uint32_t* __restrict__ maxkey, float* __restrict__ sumexp) {
    int d = blockIdx.x * blockDim.x + threadIdx.x;
    if (d == 0) { *maxkey = 0u; *sumexp = 0.0f; }
    if (d < DD) {
        float s = 0.0f;
        for (int b = 0; b < NB; ++b) s += freq[b * DD + d];
        float ls = logf(s);
        for (int b = 0; b < NB; ++b) logp[b * DD + d] = logf(freq[b * DD + d]) - ls;
    }
}

// ---- K1: main kernel. Tables staged to LDS with CDNA5 async global->LDS DMA. ----
__global__ __launch_bounds__(1024) void k_logits(const float* __restrict__ inputs,
                                                 const float* __restrict__ edges,
                                                 const float* __restrict__ logp_g,
                                                 float* __restrict__ logits) {
    extern __shared__ __align__(16) float smem[];
    float* logp_lds  = smem;            // [64][512]
    float* edges_lds = smem + NB * DD;  // [65][512]
    const uint32_t lds_logp  = (uint32_t)(uintptr_t)smem;          // LDS byte offset (low 32 bits)
    const uint32_t lds_edges = lds_logp + NB * DD * 4u;

    const int tid = threadIdx.x;

    // Async copy log_probs table: 8192 x 16B chunks (GVS mode: saddr + vaddr32).
    for (int c = tid; c < (NB * DD) / 4; c += 1024) {
        uint32_t l = lds_logp + (uint32_t)c * 16u;
        uint32_t off = (uint32_t)c * 16u;
        asm volatile("global_load_async_to_lds_b128 %0, %1, %2"
                     :: "v"(l), "v"(off), "s"(logp_g) : "memory");
    }
    // Async copy edges table: 8320 x 16B chunks.
    for (int c = tid; c < ((NB + 1) * DD) / 4; c += 1024) {
        uint32_t l = lds_edges + (uint32_t)c * 16u;
        uint32_t off = (uint32_t)c * 16u;
        asm volatile("global_load_async_to_lds_b128 %0, %1, %2"
                     :: "v"(l), "v"(off), "s"(edges) : "memory");
    }
    asm volatile("s_wait_asynccnt 0" ::: "memory");
    __syncthreads();

    const int lane = tid & 31;
    const int wv   = tid >> 5;                       // 0..31 waves per block
    const int gw   = blockIdx.x * 32 + wv;           // global wave id
    const int nw   = gridDim.x * 32;                 // total waves

    const float* ecol = edges_lds + lane;            // bank = lane (conflict-free)
    const float* pcol = logp_lds + lane;

    for (int n = gw; n < NN; n += nw) {
        const float* row = inputs + (size_t)n * DD;
        float acc = 0.0f;
#pragma unroll 4
        for (int k = 0; k < DD / 32; ++k) {          // 16 features per lane
            const int d = (k << 5);                  // + lane via ecol/pcol base
            const float x = row[lane + d];
            // branchless binary search over edges[1..64] for feature (lane + d):
            // count = #{ j in 1..64 : edges[j] <= x }  ==  searchsorted(right)-1 pre-clip
            int base = 0;
#pragma unroll
            for (int s = 32; s >= 1; s >>= 1) {
                if (ecol[(base + s) * DD + d] <= x) base += s;
            }
            const int bin = base > 63 ? 63 : base;   // clip (count==64 -> 63)
            acc += pcol[bin * DD + d];
        }
        // wave32 reduction over the 32 feature-partials
        for (int off = 16; off >= 1; off >>= 1) acc += __shfl_xor(acc, off, 32);
        if (lane == 0) logits[n] = acc;
    }
}

// ---- K2: global max of logits (order-independent => deterministic) ----
__global__ void k_max(const float* __restrict__ logits, uint32_t* __restrict__ maxkey) {
    int i = blockIdx.x * blockDim.x + threadIdx.x;
    int stride = gridDim.x * blockDim.x;
    float m = -3.4e38f;
    for (int n = i; n < NN; n += stride) m = fmaxf(m, logits[n]);
    for (int off = 16; off >= 1; off >>= 1) m = fmaxf(m, __shfl_xor(m, off, 32));
    if ((threadIdx.x & 31) == 0) atomicMax(maxkey, fkey(m));
}

// ---- K3: per-block partial sums of exp(l - max) (deterministic tree) ----
__global__ void k_sumexp(const float* __restrict__ logits, const uint32_t* __restrict__ maxkey,
                         float* __restrict__ partials) {
    __shared__ float wsum[8];
    const float mx = funkey(*maxkey);
    int i = blockIdx.x * blockDim.x + threadIdx.x;
    int stride = gridDim.x * blockDim.x;
    float s = 0.0f;
    for (int n = i; n < NN; n += stride) s += expf(logits[n] - mx);
    for (int off = 16; off >= 1; off >>= 1) s += __shfl_xor(s, off, 32);
    if ((threadIdx.x & 31) == 0) wsum[threadIdx.x >> 5] = s;
    __syncthreads();
    if (threadIdx.x == 0) {
        float t = 0.0f;
        for (int w = 0; w < 8; ++w) t += wsum[w];
        partials[blockIdx.x] = t;
    }
}

// ---- K4: reduce 64 partials with a single wave ----
__global__ void k_finalize(const float* __restrict__ partials, float* __restrict__ sumexp) {
    int lane = threadIdx.x;
    float s = partials[lane] + partials[lane + 32];
    for (int off = 16; off >= 1; off >>= 1) s += __shfl_xor(s, off, 32);
    if (lane == 0) *sumexp = s;
}

// ---- K5: out[n] = exp(l - max) / sumexp (in place on d_out) ----
__global__ void k_norm(float* __restrict__ out, const uint32_t* __restrict__ maxkey,
                       const float* __restrict__ sumexp) {
    int n = blockIdx.x * blockDim.x + threadIdx.x;
    if (n < NN) {
        const float mx = funkey(*maxkey);
        const float inv = 1.0f / *sumexp;
        out[n] = expf(out[n] - mx) * inv;
    }
}

extern "C" void kernel_launch(void* const* d_in, const int* in_sizes, int n_in,
                              void* d_out, int out_size, void* d_ws, size_t ws_size,
                              hipStream_t stream) {
    const float* inputs = (const float*)d_in[0];   // [131072, 512]
    const float* freq   = (const float*)d_in[1];   // [64, 512]
    const float* edges  = (const float*)d_in[2];   // [65, 512]
    float* out = (float*)d_out;                    // [131072] f32

    // workspace layout: logp[64*512] | maxkey(u32) | sumexp(f32) | partials[64]
    float*    logp     = (float*)d_ws;
    uint32_t* maxkey   = (uint32_t*)((char*)d_ws + (size_t)NB * DD * 4);
    float*    sumexp   = (float*)((char*)d_ws + (size_t)NB * DD * 4 + 4);
    float*    partials = (float*)((char*)d_ws + (size_t)NB * DD * 4 + 8);

    k_logp<<<2, 256, 0, stream>>>(freq, logp, maxkey, sumexp);

    const size_t shmem = (size_t)(NB + NB + 1) * DD * sizeof(float); // 264192 B <= 320KB WGP LDS
    k_logits<<<256, 1024, shmem, stream>>>(inputs, edges, logp, out);

    k_max<<<64, 256, 0, stream>>>(out, maxkey);
    k_sumexp<<<64, 256, 0, stream>>>(out, maxkey, partials);
    k_finalize<<<1, 32, 0, stream>>>(partials, sumexp);
    k_norm<<<NN / 256, 256, 0, stream>>>(out, maxkey, sumexp);
}